// HashEncoder_84198538871546
// MI455X (gfx1250) — compile-verified
//
#include <hip/hip_runtime.h>
#include <math.h>

#define H_LEVELS 16
#define BEGIN_FAST 5        // provably: level-5 res=81, 81^3=531441 > 2^19 (first hashed level)
#define HASH_P1 2654435761u
#define HASH_P2 805459861u

typedef float v2f __attribute__((ext_vector_type(2)));
typedef float v4f __attribute__((ext_vector_type(4)));
typedef float v4f_a8 __attribute__((ext_vector_type(4), aligned(8)));  // 8B-aligned pair of float2

struct LevelMeta {
  float    scale[H_LEVELS];
  unsigned offset[H_LEVELS];
  unsigned res[H_LEVELS];
};

// LDS-resident portion: level 0 (4096 entries) + level 1 (12168 entries), contiguous
// from the start of the table. 16264 entries * 8B = 130112B; pad to 16384 entries
// (131072B = 16 * 512 threads * 16B) so async staging needs no tail predication.
#define LDS_ENTRIES 16384

__global__ __launch_bounds__(512) void hashenc_kernel(
    const float* __restrict__ pos,
    const float* __restrict__ table,
    float* __restrict__ out,
    int B, LevelMeta meta)
{
#pragma clang fp contract(off)
  __shared__ __attribute__((aligned(16))) v2f ldsT[LDS_ENTRIES];  // 128 KiB

  const unsigned t = threadIdx.x;

  // ---- Stage levels 0..1 of the table into LDS with CDNA5 async copies ----
  {
    unsigned lbase = (unsigned)(unsigned long long)(&ldsT[0]); // low 32 bits = LDS offset
    unsigned long long gbase = (unsigned long long)table;      // levels 0,1 start at offset 0
#pragma unroll
    for (int j = 0; j < 16; ++j) {
      unsigned byteoff = t * 16u + (unsigned)j * 8192u;        // 512 thr * 16B = 8KB per pass
      unsigned laddr = lbase + byteoff;
      unsigned long long gaddr = gbase + byteoff;
      asm volatile("global_load_async_to_lds_b128 %0, %1, off"
                   :: "v"(laddr), "v"(gaddr) : "memory");
    }
    asm volatile("s_wait_asynccnt 0" ::: "memory");
  }
  __syncthreads();

  const v2f* __restrict__ tb = (const v2f*)table;
  const int stride = (int)(blockDim.x * gridDim.x);

  for (int i = (int)(blockIdx.x * blockDim.x + t); i < B; i += stride) {
    const float px = __builtin_nontemporal_load(pos + 3 * (size_t)i + 0);
    const float py = __builtin_nontemporal_load(pos + 3 * (size_t)i + 1);
    const float pz = __builtin_nontemporal_load(pos + 3 * (size_t)i + 2);
    float* op = out + 32 * (size_t)i;

    float p0 = 0.0f, p1 = 0.0f;  // buffered even-level result for paired b128 store

#pragma unroll
    for (int L = 0; L < H_LEVELS; ++L) {
      const float scale = (L == 0) ? 15.0f : meta.scale[L];
      float fx = px * scale + 0.5f, fy = py * scale + 0.5f, fz = pz * scale + 0.5f;
      float gx = floorf(fx), gy = floorf(fy), gz = floorf(fz);
      float rx = fx - gx, ry = fy - gy, rz = fz - gz;
      float ox = 1.0f - rx, oy = 1.0f - ry, oz = 1.0f - rz;
      unsigned ux = (unsigned)gx, uy = (unsigned)gy, uz = (unsigned)gz;

      float a0 = 0.0f, a1 = 0.0f;
      unsigned hx0 = ux, hx1 = ux + 1u;

      if (L <= 1) {
        // ---- dense levels 0..1, tables resident in LDS (ds_load / 2addr forms) ----
        const unsigned rr = (L == 0) ? 16u : meta.res[1];
        const unsigned rr2 = rr * rr;
        const unsigned base = (L == 0) ? 0u : 4096u;   // offset[1] == size[0] == 4096
        unsigned hy0 = uy * rr,  hy1 = hy0 + rr;
        unsigned hz0 = uz * rr2, hz1 = hz0 + rr2;
#pragma unroll
        for (int c = 0; c < 8; ++c) {
          unsigned h = ((c & 1) ? hx1 : hx0) + ((c & 2) ? hy1 : hy0) + ((c & 4) ? hz1 : hz0);
          // only level 0 (integer scale 15.0) can wrap; its size 4096 is pow2
          unsigned idx = (L == 0) ? (h & 4095u) : (base + h);
          float w = (((c & 1) ? rx : ox) * ((c & 2) ? ry : oy)) * ((c & 4) ? rz : oz);
          v2f f = ldsT[idx];
          a0 = a0 + w * f.x;
          a1 = a1 + w * f.y;
        }
      } else if (L < BEGIN_FAST) {
        // ---- dense levels 2..4: h < size proven, and the ex=0/ex=1 corner pair is
        //      contiguous (h, h+1) -> fetch both float2s with one 16B load ----
        const v2f* __restrict__ tbL = tb + meta.offset[L];
        const unsigned rr = meta.res[L];
        const unsigned rr2 = rr * rr;
        unsigned hy0 = uy * rr,  hy1 = hy0 + rr;
        unsigned hz0 = uz * rr2, hz1 = hz0 + rr2;
#pragma unroll
        for (int c = 0; c < 8; c += 2) {
          unsigned idx = hx0 + ((c & 2) ? hy1 : hy0) + ((c & 4) ? hz1 : hz0);
          float wy = (c & 2) ? ry : oy;
          float wz = (c & 4) ? rz : oz;
          v4f_a8 f = *(const v4f_a8*)(tbL + idx);      // entries idx (ex=0) and idx+1 (ex=1)
          float w0 = (ox * wy) * wz;                    // exact original association & order
          float w1 = (rx * wy) * wz;
          a0 = a0 + w0 * f.x;
          a1 = a1 + w0 * f.y;
          a0 = a0 + w1 * f.z;
          a1 = a1 + w1 * f.w;
        }
      } else {
        // ---- hashed levels 5..15: size is always 2^19, scattered 8B gathers (L2) ----
        const v2f* __restrict__ tbL = tb + meta.offset[L];
        unsigned hy0 = uy * HASH_P1, hy1 = hy0 + HASH_P1;  // (uy+1)*P1 mod 2^32
        unsigned hz0 = uz * HASH_P2, hz1 = hz0 + HASH_P2;
#pragma unroll
        for (int c = 0; c < 8; ++c) {
          unsigned h = ((c & 1) ? hx1 : hx0) ^ ((c & 2) ? hy1 : hy0) ^ ((c & 4) ? hz1 : hz0);
          unsigned idx = h & 0x7FFFFu;
          float w = (((c & 1) ? rx : ox) * ((c & 2) ? ry : oy)) * ((c & 4) ? rz : oz);
          v2f f = tbL[idx];
          a0 = a0 + w * f.x;
          a1 = a1 + w * f.y;
        }
      }

      if (L & 1) {
        v4f r; r.x = p0; r.y = p1; r.z = a0; r.w = a1;
        __builtin_nontemporal_store(r, (v4f*)(op + 2 * (L - 1)));  // streamed, bypass L2
      } else {
        p0 = a0; p1 = a1;
      }
    }
  }
}

extern "C" void kernel_launch(void* const* d_in, const int* in_sizes, int n_in,
                              void* d_out, int out_size, void* d_ws, size_t ws_size,
                              hipStream_t stream) {
  const float* positions = (const float*)d_in[0];
  const float* table     = (const float*)d_in[1];
  float* out             = (float*)d_out;
  const int B = in_sizes[0] / 3;

  // Replay HashEncoder.__init__ bookkeeping in double precision (matches numpy).
  LevelMeta meta;
  const double LOG_B = log(2048.0 / 16.0) / 15.0;
  long long offset = 0;
  for (int i = 0; i < H_LEVELS; ++i) {
    double s = 16.0 * exp((double)i * LOG_B) - 1.0;        // grid_scale
    long long res = (long long)(ceil(s) + 1.0);            // grid_resolution
    double full = (double)res * (double)res * (double)res;
    long long aligned = (long long)((full + 7.0) / 8.0) * 8ll;
    long long size = aligned < 524288ll ? aligned : 524288ll;
    meta.scale[i]  = (float)s;
    meta.offset[i] = (unsigned)offset;
    meta.res[i]    = (unsigned)res;
    offset += size;
  }

  const int threads = 512;
  int maxb = (B + threads - 1) / threads;
  int blocks = maxb < 2048 ? maxb : 2048;   // grid-stride: amortize LDS staging
  if (blocks < 1) blocks = 1;
  hipLaunchKernelGGL(hashenc_kernel, dim3(blocks), dim3(threads), 0, stream,
                     positions, table, out, B, meta);
}